// SPUTrainable_60352880444030
// MI455X (gfx1250) — compile-verified
//
#include <hip/hip_runtime.h>
#include <cstdint>

// ---------------------------------------------------------------------------
// SPU bound propagation (RIAI SPUTrainable forward), elementwise on (bs, n).
// out[i] = (al, au) per element. Inputs staged into LDS via gfx1250 async
// DMA loads (global_load_async_to_lds_b32, ASYNCcnt-tracked).
// The 20-step bisection (the VALU hot loop) is specialized for c < 0 and
// shares a single native exp per step; everything else uses accurate math.
// ---------------------------------------------------------------------------

__device__ __forceinline__ float sigm(float x) {
    return 1.0f / (1.0f + expf(-x));
}

// spu(x) = x>=0 ? x^2 - 0.5 : sigmoid(-x) - 1
__device__ __forceinline__ float spu(float x) {
    float pos = fmaf(x, x, -0.5f);
    float neg = 1.0f / (1.0f + expf(x)) - 1.0f;   // sigmoid(-x) - 1
    return (x >= 0.0f) ? pos : neg;
}

// spu'(x): x<=-50 -> 0 ; x>=0 -> 2x ; else -0.5/(cosh(x)+1)
__device__ __forceinline__ float spu_grad(float x) {
    float mid = -0.5f / (coshf(x) + 1.0f);
    float r   = (x >= 0.0f) ? (2.0f * x) : mid;
    return (x <= -50.0f) ? 0.0f : r;
}

__device__ __forceinline__ void spu_tangent(float x0, float& a, float& b) {
    a = spu_grad(x0);
    b = spu(x0) - a * x0;
}

__device__ __forceinline__ void tangent_param(float l, float u, float t,
                                              float& a, float& b) {
    spu_tangent(l * (1.0f - t) + u * t, a, b);
}

__device__ __forceinline__ float spu_shifted_over_x(float x) {
    bool  normal = (x <= -1e-5f) || (x > 0.0f);
    float sx     = normal ? x : 1.0f;
    float v      = (spu(x) + 0.5f) / sx;
    return normal ? v : -0.25f;
}

// Hot loop. For all lanes whose bisection result is consumed (crossing:
// l < 0 < u), c = (bl+br)/2 < 0 strictly, so:
//   spu(c)      = -e/(1+e)            (sigmoid branch, e = exp(c) < 1)
//   spu_grad(c) = -0.5/(cosh(c)+1),  cosh(c) = (e + 1/e)/2
// One native exp + three v_rcp per step instead of OCML coshf+expf+2 divs.
// Non-crossing lanes produce select-discarded garbage, same as the
// reference's masked-out lanes.
__device__ __forceinline__ float binsearch_acceptable_range(float l, float u) {
    float req = spu(u);
    float bl = l, br = 0.0f;
#pragma unroll
    for (int i = 0; i < 20; ++i) {
        float c  = 0.5f * (bl + br);
        float e  = __expf(c);                                   // v_exp_f32
        float ch = 0.5f * (e + __builtin_amdgcn_rcpf(e));       // cosh(c)
        float a  = -0.5f * __builtin_amdgcn_rcpf(ch + 1.0f);    // spu'(c)
        a        = (c <= -50.0f) ? 0.0f : a;
        float sp = -e * __builtin_amdgcn_rcpf(1.0f + e);        // spu(c)
        float b  = sp - a * c;
        bool covers = (a * u + b) >= req;
        bl = covers ? c : bl;
        br = covers ? br : c;
    }
    return bl;
}

// gfx1250 async DMA: copy one dword per lane from global to LDS.
// Generic LDS pointer low 32 bits == hardware LDS byte address (aperture map).
__device__ __forceinline__ void async_load_b32(void* lds_dst, const void* gbase,
                                               unsigned voff_bytes) {
    unsigned loff = (unsigned)(uintptr_t)lds_dst;
    asm volatile("global_load_async_to_lds_b32 %0, %1, %2"
                 :
                 : "v"(loff), "v"(voff_bytes), "s"(gbase)
                 : "memory");
}

__device__ __forceinline__ void wait_asynccnt0() {
    asm volatile("s_wait_asynccnt 0" ::: "memory");
}

#define BLOCK 256

__global__ __launch_bounds__(BLOCK)
void spu_forward_kernel(const float* __restrict__ gl,
                        const float* __restrict__ gu,
                        const float* __restrict__ gtl,
                        const float* __restrict__ gtu,
                        float2* __restrict__ out, int total) {
    __shared__ float sl[BLOCK], su[BLOCK], stl[BLOCK], stu[BLOCK];

    const int t   = threadIdx.x;
    const int gid = blockIdx.x * BLOCK + t;
    const int cid = (gid < total) ? gid : (total - 1);   // clamp, store-guarded
    const unsigned voff = (unsigned)cid * 4u;

    // Stage inputs through LDS with async DMA (ASYNCcnt-tracked).
    async_load_b32(&sl[t],  gl,  voff);
    async_load_b32(&su[t],  gu,  voff);
    async_load_b32(&stl[t], gtl, voff);
    async_load_b32(&stu[t], gtu, voff);
    wait_asynccnt0();   // each lane consumes only its own slots -> no barrier

    const float l  = sl[t];
    const float u  = su[t];
    const float tl = sigm(4.0f * stl[t]);
    const float tu = sigm(4.0f * stu[t]);

    // ---- endpoint (secant) line ----
    const float yl = spu(l), yu = spu(u);
    const float ea = (yu - yl) / (u - l + 1e-8f);
    const float eb = yl - l * ea;

    const bool negative = (u <= 0.0f);
    const bool positive = (l >= 0.0f);
    const bool crossing = !(negative || positive);

    // ---- negative branch (masked args) ----
    const float ln = negative ? l : 0.0f, un = negative ? u : 0.0f;
    float na, nb;
    tangent_param(ln, un, tu, na, nb);

    // ---- positive branch (masked args) ----
    const float lp = positive ? l : 0.0f, up = positive ? u : 0.0f;
    float pa, pb;
    tangent_param(lp, up, tl, pa, pb);

    // ---- crossing branch: upper bound ----
    const float uc    = crossing ? u : 0.0f;
    const float u_end = binsearch_acceptable_range(l, uc);
    float ua, ub2;
    tangent_param(l, u_end, tu, ua, ub2);
    const bool  not_cov = (ua * uc + ub2) <= spu(uc);
    const float cr_uw = not_cov ? ea : ua;
    const float cr_ub = not_cov ? eb : ub2;

    // ---- crossing branch: lower bound ----
    const float den       = uc - l;
    const float threshold = -l / ((den == 0.0f) ? 1.0f : den);
    const bool  is_normal = (tl >= threshold);
    float nla, nlb;
    tangent_param(l, uc, tl, nla, nlb);
    const float steep    = spu_shifted_over_x(l);
    const float safe_thr = (threshold == 0.0f) ? 1.0f : threshold;
    const float rr       = tl / safe_thr;
    const float wa       = steep * (1.0f - rr * rr);
    const float wb       = -0.5f;
    const float cr_lw = is_normal ? nla : wa;
    const float cr_lb = is_normal ? nlb : wb;

    // ---- combine (predicates are mutually exclusive) ----
    const float lw = negative ? ea : (positive ? pa : cr_lw);
    const float lb = negative ? eb : (positive ? pb : cr_lb);
    const float uw = negative ? na : (positive ? ea : cr_uw);
    const float ub = negative ? nb : (positive ? eb : cr_ub);

    // ---- backsubstitute into input box (diagonal matmul) ----
    const float al = fmaxf(lw, 0.0f) * l + fminf(lw, 0.0f) * u + lb;
    const float au = fmaxf(uw, 0.0f) * u + fminf(uw, 0.0f) * l + ub;

    if (gid < total) {
        float2 r;
        r.x = al;
        r.y = au;
        out[gid] = r;
    }
}

extern "C" void kernel_launch(void* const* d_in, const int* in_sizes, int n_in,
                              void* d_out, int out_size, void* d_ws, size_t ws_size,
                              hipStream_t stream) {
    const float* l  = (const float*)d_in[0];
    const float* u  = (const float*)d_in[1];
    const float* tl = (const float*)d_in[2];
    const float* tu = (const float*)d_in[3];
    const int total = in_sizes[0];          // bs * n elements
    const int grid  = (total + BLOCK - 1) / BLOCK;
    spu_forward_kernel<<<grid, BLOCK, 0, stream>>>(l, u, tl, tu,
                                                   (float2*)d_out, total);
}